// SimLoss_15161234555302
// MI455X (gfx1250) — compile-verified
//
#include <hip/hip_runtime.h>
#include <stdint.h>

#define HW     512
#define LPIX   (HW * HW)        // 2^18
#define PADW   4
#define CHUNKS (LPIX / 256)     // 1024 chunks of 256 elements per row

typedef float v2f __attribute__((ext_vector_type(2)));
typedef float v8f __attribute__((ext_vector_type(8)));

#if __has_builtin(__builtin_amdgcn_ballot_w32)
#define BALLOT32(p) __builtin_amdgcn_ballot_w32((bool)(p))
#else
#define BALLOT32(p) ((uint32_t)__ballot((int)(p)))
#endif

__device__ __forceinline__ int refl_idx(int p) {
  p = (p < 0) ? -p : p;
  p = (p >= HW) ? (2 * HW - 2 - p) : p;
  return p;
}

// ---------------- Morphology (exactly reproduces the blur thresholds) ----------------
// hbuf bit0 = OR over horizontal 9-window of t, bit1 = AND, bit2 = t (center).
__global__ void k_morph_h(const int* __restrict__ tgt, uint8_t* __restrict__ hbuf, int total) {
  int idx = blockIdx.x * blockDim.x + threadIdx.x;
  if (idx >= total) return;
  int img = idx >> 18;
  int rem = idx & (LPIX - 1);
  int i = rem >> 9;
  int j = rem & (HW - 1);
  const int* row = tgt + img * LPIX + i * HW;
  int o = 0, a = 1;
#pragma unroll
  for (int dj = -PADW; dj <= PADW; ++dj) {
    int v = row[refl_idx(j + dj)];
    o |= v; a &= v;
  }
  int c = row[j];
  hbuf[idx] = (uint8_t)((o & 1) | ((a & 1) << 1) | ((c & 1) << 2));
}

// mbuf bit0 = fp (9x9 dilation), bit1 = fn (9x9 erosion), bit2 = t (center)
__global__ void k_morph_v(const uint8_t* __restrict__ hbuf, uint8_t* __restrict__ mbuf, int total) {
  int idx = blockIdx.x * blockDim.x + threadIdx.x;
  if (idx >= total) return;
  int img = idx >> 18;
  int rem = idx & (LPIX - 1);
  int i = rem >> 9;
  int j = rem & (HW - 1);
  const uint8_t* base = hbuf + img * LPIX + j;
  int o = 0, a = 1;
#pragma unroll
  for (int di = -PADW; di <= PADW; ++di) {
    int h = base[refl_idx(i + di) * HW];
    o |= h;          // bit0: OR
    a &= (h >> 1);   // bit1: AND
  }
  int c = (base[i * HW] >> 2) & 1;
  mbuf[idx] = (uint8_t)((o & 1) | ((a & 1) << 1) | (c << 2));
}

// ---------------- Phase 1: per-chunk mask counts (fully parallel) ----------------
// One block = one 256-element chunk. Counts both fallback variants so no serialization.
__global__ void k_cnt(const float* __restrict__ x, const uint8_t* __restrict__ mbuf,
                      unsigned* __restrict__ Tpos, unsigned* __restrict__ TfpA,
                      unsigned* __restrict__ TfpB, unsigned* __restrict__ TfnA,
                      unsigned* __restrict__ TfnB) {
  const int g   = blockIdx.x;             // global chunk id
  const int tid = threadIdx.x;
  const int lane = tid & 31, wid = tid >> 5;
  const int idx = g * 256 + tid;

  float   xv = x[idx];
  uint8_t m  = mbuf[idx];
  int fp = m & 1, fn = (m >> 1) & 1, tv = (m >> 2) & 1;

  uint32_t b0 = BALLOT32(tv != 0);
  uint32_t b1 = BALLOT32((xv > 0.f) && (fp == 0));
  uint32_t b2 = BALLOT32(fp == 0);
  uint32_t b3 = BALLOT32((xv < 0.f) && (fn != 0));
  uint32_t b4 = BALLOT32(fn != 0);

  __shared__ unsigned w[5][8];
  if (lane == 0) {
    w[0][wid] = (unsigned)__popc(b0);
    w[1][wid] = (unsigned)__popc(b1);
    w[2][wid] = (unsigned)__popc(b2);
    w[3][wid] = (unsigned)__popc(b3);
    w[4][wid] = (unsigned)__popc(b4);
  }
  __syncthreads();
  if (tid < 5) {
    unsigned s = 0;
#pragma unroll
    for (int k = 0; k < 8; ++k) s += w[tid][k];
    unsigned* T[5] = {Tpos, TfpA, TfpB, TfnA, TfnB};
    T[tid][g] = s;
  }
}

// ---------------- Phase 2: per-row scan of chunk counts (32 small blocks) ----------------
__device__ __forceinline__ unsigned scan_excl_1024(unsigned v, unsigned* sh, unsigned* total) {
  const int t = threadIdx.x;
  sh[t] = v;
  __syncthreads();
  for (int d = 1; d < 1024; d <<= 1) {
    unsigned add = (t >= d) ? sh[t - d] : 0u;
    __syncthreads();
    sh[t] += add;
    __syncthreads();
  }
  unsigned tot = sh[1023];
  unsigned incl = sh[t];
  __syncthreads();          // safe to reuse sh afterwards
  *total = tot;
  return incl - v;
}

__global__ void k_scan(const unsigned* __restrict__ Tpos, const unsigned* __restrict__ TfpA,
                       const unsigned* __restrict__ TfpB, const unsigned* __restrict__ TfnA,
                       const unsigned* __restrict__ TfnB,
                       unsigned* __restrict__ CBpos, unsigned* __restrict__ CBfp,
                       unsigned* __restrict__ CBfn, int* __restrict__ cnt) {
  const int n = blockIdx.x;
  const int t = threadIdx.x;
  const int g = n * CHUNKS + t;
  __shared__ unsigned sh[1024];

  unsigned vpos = Tpos[g], vfpA = TfpA[g], vfpB = TfpB[g], vfnA = TfnA[g], vfnB = TfnB[g];
  unsigned kpos, kfp, kneg, kfn, kp2;
  unsigned bpos  = scan_excl_1024(vpos, sh, &kpos);
  unsigned bfpA  = scan_excl_1024(vfpA, sh, &kfp);
  unsigned bfpB  = scan_excl_1024(vfpB, sh, &kneg);
  unsigned bfnA  = scan_excl_1024(vfnA, sh, &kfn);
  unsigned bfnB  = scan_excl_1024(vfnB, sh, &kp2);

  CBpos[g] = bpos;
  CBfp[g]  = (kfp > 0) ? bfpA : bfpB;
  CBfn[g]  = (kfn > 0) ? bfnA : bfnB;

  if (t == 0) {
    cnt[n * 8 + 0] = (int)kpos;
    cnt[n * 8 + 1] = (int)kneg;
    cnt[n * 8 + 2] = (int)kfp;
    cnt[n * 8 + 3] = (int)kp2;
    cnt[n * 8 + 4] = (int)kfn;
    cnt[n * 8 + 5] = (kfp > 0) ? 1 : 0;
    cnt[n * 8 + 6] = (kfn > 0) ? 1 : 0;
  }
}

// ---------------- Phase 3: fully parallel stable scatter ----------------
__global__ void k_scatter(const float* __restrict__ x, const uint8_t* __restrict__ mbuf,
                          const int* __restrict__ cnt,
                          const unsigned* __restrict__ CBpos, const unsigned* __restrict__ CBfp,
                          const unsigned* __restrict__ CBfn,
                          float* __restrict__ Cpos, float* __restrict__ Cfp,
                          float* __restrict__ Cfn) {
  const int g   = blockIdx.x;
  const int row = g / CHUNKS;
  const int tid = threadIdx.x;
  const int lane = tid & 31, wid = tid >> 5;
  const uint32_t lm = (1u << lane) - 1u;
  const int idx = g * 256 + tid;
  const int rowbase = row * LPIX;

  const bool usefp = cnt[row * 8 + 5] != 0;
  const bool usefn = cnt[row * 8 + 6] != 0;

  float   xv = x[idx];
  uint8_t m  = mbuf[idx];
  int fp = m & 1, fn = (m >> 1) & 1, tv = (m >> 2) & 1;

  uint32_t bl0 = BALLOT32(tv != 0);
  uint32_t bl1 = BALLOT32(usefp ? ((xv > 0.f) && (fp == 0)) : (fp == 0));
  uint32_t bl2 = BALLOT32(usefn ? ((xv < 0.f) && (fn != 0)) : (fn != 0));

  __shared__ unsigned w[3][8];
  if (lane == 0) {
    w[0][wid] = (unsigned)__popc(bl0);
    w[1][wid] = (unsigned)__popc(bl1);
    w[2][wid] = (unsigned)__popc(bl2);
  }
  __syncthreads();
  if (tid < 3) {
    unsigned run = 0;
#pragma unroll
    for (int k = 0; k < 8; ++k) { unsigned c = w[tid][k]; w[tid][k] = run; run += c; }
  }
  __syncthreads();

  unsigned o0 = CBpos[g] + w[0][wid] + (unsigned)__popc(bl0 & lm);
  unsigned o1 = CBfp[g]  + w[1][wid] + (unsigned)__popc(bl1 & lm);
  unsigned o2 = CBfn[g]  + w[2][wid] + (unsigned)__popc(bl2 & lm);
  if ((bl0 >> lane) & 1) Cpos[rowbase + o0] = xv;
  if ((bl1 >> lane) & 1) Cfp [rowbase + o1] = xv;
  if ((bl2 >> lane) & 1) Cfn [rowbase + o2] = xv;
}

// ---------------- Fused loss + WMMA block reduction ----------------
__device__ __forceinline__ float bce_elem(float v, float y) {
  return fmaxf(v, 0.f) - v * y + log1pf(__expf(-fabsf(v)));
}

__global__ void k_loss(const float* __restrict__ x, const uint8_t* __restrict__ mbuf,
                       const int* __restrict__ cnt,
                       const float* __restrict__ Cpos, const float* __restrict__ Cfp,
                       const float* __restrict__ Cfn, double* __restrict__ partials) {
  const int n = blockIdx.y;
  const unsigned i = blockIdx.x * 256u + threadIdx.x;
  const int base = n * LPIX;
  const int kpos = cnt[n * 8 + 0];
  const int kneg = cnt[n * 8 + 1];
  const int kfp  = cnt[n * 8 + 2];
  const int kp2  = cnt[n * 8 + 3];
  const int kfn  = cnt[n * 8 + 4];

  float xv = x[base + i];
  float tv = (float)((mbuf[base + i] >> 2) & 1);

  float dpos = 5.0f;
  if (kpos > 0) dpos = Cpos[base + (i % (unsigned)kpos)];
  float dfp = -5.0f;
  if (kneg > 0) { unsigned k = (unsigned)(kfp > 0 ? kfp : kneg); dfp = Cfp[base + (i % k)]; }
  float dfn = 5.0f;
  if (kp2 > 0)  { unsigned k = (unsigned)(kfn > 0 ? kfn : kp2);  dfn = Cfn[base + (i % k)]; }

  float s = bce_elem(dpos * xv, tv) + bce_elem(dpos, 1.f)
          + 0.6f * (bce_elem(dpos * dfp, 0.f) + bce_elem(dpos * dfn, 1.f));

  __shared__ float red[128];
  const int lane = threadIdx.x & 31;
  const int wid  = threadIdx.x >> 5;

#if __has_builtin(__builtin_amdgcn_wmma_f32_16x16x4_f32)
  // Exact fp32 wave reduction: A = ones(16x4), B = {s, 0} -> D[m][n] = s[n] + s[n+16]
  v2f a; a.x = 1.f; a.y = 1.f;
  v2f b; b.x = s;   b.y = 0.f;
  v8f c = {};
  c = __builtin_amdgcn_wmma_f32_16x16x4_f32(false, a, false, b, (short)0, c, false, false);
  if (lane < 16) red[wid * 16 + lane] = c[0];
  __syncthreads();
  if (threadIdx.x == 0) {
    float tot = 0.f;
    for (int k = 0; k < 128; ++k) tot += red[k];
    partials[(size_t)n * gridDim.x + blockIdx.x] = (double)tot;
  }
#else
  __shared__ float all[256];
  all[threadIdx.x] = s;
  __syncthreads();
  if (threadIdx.x == 0) {
    float tot = 0.f;
    for (int k = 0; k < 256; ++k) tot += all[k];
    partials[(size_t)n * gridDim.x + blockIdx.x] = (double)tot;
  }
  (void)red; (void)lane; (void)wid;
#endif
}

// ---------------- Deterministic final reduction ----------------
__global__ void k_final(const double* __restrict__ partials, int np,
                        float* __restrict__ out, double inv) {
  __shared__ double sd[256];
  double s = 0.0;
  for (int k = threadIdx.x; k < np; k += 256) s += partials[k];
  sd[threadIdx.x] = s;
  __syncthreads();
  for (int st = 128; st > 0; st >>= 1) {
    if (threadIdx.x < st) sd[threadIdx.x] += sd[threadIdx.x + st];
    __syncthreads();
  }
  if (threadIdx.x == 0) out[0] = (float)(sd[0] * inv);
}

extern "C" void kernel_launch(void* const* d_in, const int* in_sizes, int n_in,
                              void* d_out, int out_size, void* d_ws, size_t ws_size,
                              hipStream_t stream) {
  (void)n_in; (void)out_size; (void)ws_size;
  const float* x  = (const float*)d_in[0];
  const int* tgt  = (const int*)d_in[1];
  const int total = in_sizes[0];           // B*C*H*W
  const int N     = total / LPIX;          // rows (B*C)
  const int nbx   = LPIX / 256;            // 1024
  const int nparts = nbx * N;
  const int nchunks = N * CHUNKS;          // total 256-elem chunks

  char* ws = (char*)d_ws;
  size_t off = 0;
  auto wsalloc = [&](size_t bytes) -> void* {
    void* p = ws + off;
    off = (off + bytes + 255) & ~(size_t)255;
    return p;
  };
  int*      cnt      = (int*)wsalloc((size_t)N * 8 * sizeof(int));
  double*   partials = (double*)wsalloc((size_t)nparts * sizeof(double));
  uint8_t*  hbuf     = (uint8_t*)wsalloc((size_t)total);
  uint8_t*  mbuf     = (uint8_t*)wsalloc((size_t)total);
  unsigned* Tpos     = (unsigned*)wsalloc((size_t)nchunks * sizeof(unsigned));
  unsigned* TfpA     = (unsigned*)wsalloc((size_t)nchunks * sizeof(unsigned));
  unsigned* TfpB     = (unsigned*)wsalloc((size_t)nchunks * sizeof(unsigned));
  unsigned* TfnA     = (unsigned*)wsalloc((size_t)nchunks * sizeof(unsigned));
  unsigned* TfnB     = (unsigned*)wsalloc((size_t)nchunks * sizeof(unsigned));
  unsigned* CBpos    = (unsigned*)wsalloc((size_t)nchunks * sizeof(unsigned));
  unsigned* CBfp     = (unsigned*)wsalloc((size_t)nchunks * sizeof(unsigned));
  unsigned* CBfn     = (unsigned*)wsalloc((size_t)nchunks * sizeof(unsigned));
  float*    Cpos     = (float*)wsalloc((size_t)total * sizeof(float));
  float*    Cfp      = (float*)wsalloc((size_t)total * sizeof(float));
  float*    Cfn      = (float*)wsalloc((size_t)total * sizeof(float));

  const int tb = (total + 255) / 256;
  k_morph_h<<<tb, 256, 0, stream>>>(tgt, hbuf, total);
  k_morph_v<<<tb, 256, 0, stream>>>(hbuf, mbuf, total);
  k_cnt<<<nchunks, 256, 0, stream>>>(x, mbuf, Tpos, TfpA, TfpB, TfnA, TfnB);
  k_scan<<<N, 1024, 0, stream>>>(Tpos, TfpA, TfpB, TfnA, TfnB, CBpos, CBfp, CBfn, cnt);
  k_scatter<<<nchunks, 256, 0, stream>>>(x, mbuf, cnt, CBpos, CBfp, CBfn, Cpos, Cfp, Cfn);
  k_loss<<<dim3(nbx, N), 256, 0, stream>>>(x, mbuf, cnt, Cpos, Cfp, Cfn, partials);
  k_final<<<1, 256, 0, stream>>>(partials, nparts, (float*)d_out, 1.0 / (double)total);
}